// DeepAggregateLayer_7267084665149
// MI455X (gfx1250) — compile-verified
//
#include <hip/hip_runtime.h>
#include <hip/hip_bf16.h>
#include <stdint.h>

// Problem constants (from reference)
#define B_TOT   512
#define IN_F    4096
#define OUT_F   4096
#define NCONN   64
// Tiling
#define ROWS    8                  // batch rows staged per workgroup (2 quads of 4)
#define THREADS 256                // one output feature per thread
// Quad layout: LDS holds, for quad q in {0,1}, float4 per feature idx:
//   lds_x[q*QSTRIDE + idx*4 + r]  == x[b0 + 4*q + r][idx]
// One ds_load_b128 fetches 4 batch rows of one feature.
#define QSTRIDE (IN_F * 4 + 16)    // 16400 floats; 65600 B, 16B aligned

typedef int v4i __attribute__((ext_vector_type(4)));

__device__ __forceinline__ float4 f4add(float4 a, float4 b) {
    return make_float4(a.x + b.x, a.y + b.y, a.z + b.z, a.w + b.w);
}
__device__ __forceinline__ float4 f4max(float4 a, float4 b) {
    return make_float4(fmaxf(a.x, b.x), fmaxf(a.y, b.y), fmaxf(a.z, b.z), fmaxf(a.w, b.w));
}
__device__ __forceinline__ float4 f4min(float4 a, float4 b) {
    return make_float4(fminf(a.x, b.x), fminf(a.y, b.y), fminf(a.z, b.z), fminf(a.w, b.w));
}

__global__ __launch_bounds__(THREADS)
void DeepAggregateLayer_kernel(const float* __restrict__ x,
                               const int*   __restrict__ conn,
                               const int*   __restrict__ opidx,
                               float*       __restrict__ out_fwd,   // (B, OUT)
                               float*       __restrict__ out_agg)   // (B, 4, OUT)
{
    extern __shared__ __align__(16) float lds_x[];   // QSTRIDE + IN_F*4 floats

    const int tid = threadIdx.x;
    const int o   = blockIdx.x * THREADS + tid;   // output feature
    const int b0  = blockIdx.y * ROWS;            // first batch row of tile

    // ---- Stage ROWS rows of x into LDS, interleaved 4-rows-per-float4 --------
#if __has_builtin(__builtin_amdgcn_global_load_async_to_lds_b32)
    #pragma unroll
    for (int r = 0; r < ROWS; ++r) {
        const int    q   = r >> 2;
        const int    r4  = r & 3;
        const float* src = x + (size_t)(b0 + r) * IN_F;
        float*       dst = lds_x + q * QSTRIDE + r4;
        #pragma unroll
        for (int i = 0; i < IN_F / THREADS; ++i) {   // 16 iterations
            const int j = tid + i * THREADS;         // coalesced global reads
            __builtin_amdgcn_global_load_async_to_lds_b32(
                (int*)(src + j), (int*)(dst + j * 4),
                /*offset=*/0, /*cpol=*/0);
        }
    }
  #if __has_builtin(__builtin_amdgcn_s_wait_asynccnt)
    __builtin_amdgcn_s_wait_asynccnt(0);
  #else
    asm volatile("s_wait_asynccnt 0" ::: "memory");
  #endif
#else
    #pragma unroll
    for (int r = 0; r < ROWS; ++r) {
        const int    q   = r >> 2;
        const int    r4  = r & 3;
        const float* src = x + (size_t)(b0 + r) * IN_F;
        float*       dst = lds_x + q * QSTRIDE + r4;
        #pragma unroll
        for (int i = 0; i < IN_F / THREADS; ++i) {
            const int j = tid + i * THREADS;
            dst[j * 4] = src[j];
        }
    }
#endif
    __syncthreads();

    if (o >= OUT_F) return;

    // ---- Gather-reduce 64 connections, 4 rows per ds_load_b128 ---------------
    const float4* quad0 = (const float4*)(lds_x);
    const float4* quad1 = (const float4*)(lds_x + QSTRIDE);

    const float NEG_INF = -__builtin_inff();
    const float POS_INF =  __builtin_inff();
    float4 s0  = make_float4(0.f, 0.f, 0.f, 0.f);
    float4 s1  = s0;
    float4 mx0 = make_float4(NEG_INF, NEG_INF, NEG_INF, NEG_INF);
    float4 mx1 = mx0;
    float4 mn0 = make_float4(POS_INF, POS_INF, POS_INF, POS_INF);
    float4 mn1 = mn0;

    const int4* c4 = (const int4*)(conn + (size_t)o * NCONN);  // 256B-aligned
    #pragma unroll 2
    for (int kk = 0; kk < NCONN / 4; ++kk) {
        const int4 c = c4[kk];
        int idx[4] = { c.x, c.y, c.z, c.w };
        #pragma unroll
        for (int u = 0; u < 4; ++u) {
            const float4 v0 = quad0[idx[u]];   // ds_load_b128: rows 0..3
            const float4 v1 = quad1[idx[u]];   // ds_load_b128: rows 4..7
            s0  = f4add(s0, v0);   s1  = f4add(s1, v1);
            mx0 = f4max(mx0, v0);  mx1 = f4max(mx1, v1);
            mn0 = f4min(mn0, v0);  mn1 = f4min(mn1, v1);
        }
    }

    // ---- Write outputs (coalesced across lanes) ------------------------------
    const int   op  = opidx[o];
    const float inv = 1.0f / (float)NCONN;   // 2^-6, exact

    auto emit = [&](int r, float sum, float vmax, float vmin) {
        const int   b    = b0 + r;
        const float mean = sum * inv;
        const float f = (op == 0) ? mean : (op == 1) ? sum : (op == 2) ? vmax : vmin;
        out_fwd[(size_t)b * OUT_F + o] = f;
        float* agg = out_agg + ((size_t)b * 4) * OUT_F + o;
        agg[0 * OUT_F] = mean;
        agg[1 * OUT_F] = sum;
        agg[2 * OUT_F] = vmax;
        agg[3 * OUT_F] = vmin;
    };

    emit(0, s0.x, mx0.x, mn0.x);
    emit(1, s0.y, mx0.y, mn0.y);
    emit(2, s0.z, mx0.z, mn0.z);
    emit(3, s0.w, mx0.w, mn0.w);
    emit(4, s1.x, mx1.x, mn1.x);
    emit(5, s1.y, mx1.y, mn1.y);
    emit(6, s1.z, mx1.z, mn1.z);
    emit(7, s1.w, mx1.w, mn1.w);
}

extern "C" void kernel_launch(void* const* d_in, const int* in_sizes, int n_in,
                              void* d_out, int out_size, void* d_ws, size_t ws_size,
                              hipStream_t stream)
{
    (void)in_sizes; (void)n_in; (void)out_size; (void)d_ws; (void)ws_size;

    const float* x     = (const float*)d_in[0];   // (512, 4096) f32
    const int*   conn  = (const int*)  d_in[1];   // (4096, 64)  i32
    const int*   opidx = (const int*)  d_in[2];   // (4096,)     i32

    float* out     = (float*)d_out;
    float* out_fwd = out;                                  // (512, 4096)
    float* out_agg = out + (size_t)B_TOT * OUT_F;          // (512, 4, 4096)

    dim3 grid(OUT_F / THREADS, B_TOT / ROWS);              // (16, 64)
    size_t shmem = (size_t)(QSTRIDE + IN_F * 4) * sizeof(float);  // 131136 B

    DeepAggregateLayer_kernel<<<grid, dim3(THREADS), shmem, stream>>>(
        x, conn, opidx, out_fwd, out_agg);
}